// FasterRCNNPredictor_42494406427383
// MI455X (gfx1250) — compile-verified
//
#include <hip/hip_runtime.h>
#include <math.h>

// ---------------------------------------------------------------------------
// FasterRCNN predictor head + per-class greedy NMS for MI455X (gfx1250).
//   out layout (floats): [BN*4 rounded][BN sigmoid][BN*4 boxes][BN*21 probs][BN keep]
// Classes partition boxes (scores finite only where argmax==class) -> 80
// independent small NMS problems, one 256-thread (8x wave32) workgroup each.
// CDNA5 paths: TDM tensor_load_to_lds (ping-pong, s_wait_tensorcnt),
// global_load_async_to_lds_b128 + s_wait_asynccnt, global_prefetch_b8.
// ---------------------------------------------------------------------------

constexpr int kC     = 21;    // classes (incl. background) — fixed by reference
constexpr int kMAXC  = 3008;  // >= N worst case (all boxes in one class)
constexpr int kChunk = 192;   // probs rows per TDM chunk (192*21*4 = 16128 B)

typedef unsigned int v4u __attribute__((ext_vector_type(4)));
typedef int          v4i __attribute__((ext_vector_type(4)));
typedef int          v8i __attribute__((ext_vector_type(8)));

// Issue one TDM 2D tile load: rows x 21 f32, row stride 21, into LDS at lds_off.
__device__ __forceinline__ void tdm_load_probs(unsigned lds_off, const float* gsrc,
                                               int rows, int remaining) {
  unsigned long long ga = (unsigned long long)(size_t)gsrc;
  // D# group 0: count=1 | lds_addr | global_addr[56:0] | type=2 ("image")
  v4u g0 = { 1u, lds_off, (unsigned)ga,
             (unsigned)((ga >> 32) & 0x1FFFFFFull) | (2u << 30) };
  unsigned td1 = (unsigned)remaining;   // tensor_dim1 = rows left (no OOB)
  // D# group 1: data_size=2 (4B) | tensor_dim0=21 | tensor_dim1 | tile 21 x rows
  //             | tensor_dim0_stride=21
  v8i g1 = { (int)0x20000u,                         // data_size = 4 bytes
             (int)(21u << 16),                      // tensor_dim0 = 21 @bit48
             (int)((td1 & 0xFFFFu) << 16),          // tensor_dim1 lo @bit80
             (int)((td1 >> 16) | (21u << 16)),      // tensor_dim1 hi | tile_dim0=21
             (int)(unsigned)rows,                   // tile_dim1 = rows
             21, 0, 0 };                            // tensor_dim0_stride = 21
  v4i z4 = {0, 0, 0, 0};
#if __clang_major__ >= 23
  v8i z8 = {0, 0, 0, 0, 0, 0, 0, 0};
  __builtin_amdgcn_tensor_load_to_lds(g0, g1, z4, z4, z8, 0);
#else
  __builtin_amdgcn_tensor_load_to_lds(g0, g1, z4, z4, 0);
#endif
}

__global__ __launch_bounds__(256) void head_kernel(
    const float* __restrict__ nms_reg, const float* __restrict__ nms_cls,
    const float* __restrict__ rcnn_reg, const float* __restrict__ rcnn_cls,
    const int* __restrict__ red_p,
    float* __restrict__ o_round, float* __restrict__ o_sig,
    float* __restrict__ o_boxes, float* __restrict__ o_probs,
    float* __restrict__ o_keep, int BN) {
  int t = blockIdx.x * blockDim.x + threadIdx.x;
  if (t >= BN) return;
  const float red = (float)red_p[0];

  float4 nr = *(const float4*)(nms_reg + (size_t)t * 4);
  float r0 = floorf(nr.x * red) / red;
  float r1 = floorf(nr.y * red) / red;
  float r2 = ceilf(nr.z * red) / red;
  float r3 = ceilf(nr.w * red) / red;
  *(float4*)(o_round + (size_t)t * 4) = make_float4(r0, r1, r2, r3);

  float4 rr = *(const float4*)(rcnn_reg + (size_t)t * 4);
  *(float4*)(o_boxes + (size_t)t * 4) =
      make_float4(rr.x + r0, rr.y + r1, rr.z + r2, rr.w + r3);

  o_sig[t] = 1.0f / (1.0f + expf(-nms_cls[t]));

  // softmax over kC logits (fully unrolled -> registers)
  const float* cl = rcnn_cls + (size_t)t * kC;
  float e[kC];
#pragma unroll
  for (int c = 0; c < kC; ++c) e[c] = cl[c];
  float m = e[0];
#pragma unroll
  for (int c = 1; c < kC; ++c) m = fmaxf(m, e[c]);
  float s = 0.f;
#pragma unroll
  for (int c = 0; c < kC; ++c) { e[c] = expf(e[c] - m); s += e[c]; }
  float* po = o_probs + (size_t)t * kC;
#pragma unroll
  for (int c = 0; c < kC; ++c) po[c] = e[c] / s;

  o_keep[t] = 0.0f;  // kernel 2 sets kept boxes to 1.0
}

// One workgroup per (batch, class>=1). LDS layout (60.6 KB, phase-aliased):
//   [0,48128)     A/B: sc[3008] f32 | nidx[3008] i32 | (+24064) TDM stage x2
//                 C/D: bbox[3008] float4 (sorted boxes, AoS); B: rk at +8*kMAXC
//   [48128,60160) sidx[3008] i32   (stage buf 1 overlaps; dead until phase B2)
//   [60160,60544) sup[96] u32 suppression bitmask
//   [60544,60548) cnt
__global__ __launch_bounds__(256) void nms_kernel(
    const float* __restrict__ boxes, const float* __restrict__ probs,
    float* __restrict__ keep, int N) {
  __shared__ alignas(16) unsigned char smem[kMAXC * 16 + kMAXC * 4 + 96 * 4 + 16];
  float*    sc   = (float*)smem;
  int*      nidx = (int*)(smem + kMAXC * 4);
  int*      rk   = (int*)(smem + kMAXC * 8);
  float4*   bbox = (float4*)smem;
  int*      sidx = (int*)(smem + kMAXC * 16);
  unsigned* sup  = (unsigned*)(smem + kMAXC * 16 + kMAXC * 4);
  int*      cnt  = (int*)(smem + kMAXC * 16 + kMAXC * 4 + 96 * 4);

  const int tid = threadIdx.x;
  const int b   = blockIdx.x / (kC - 1);
  const int cls = 1 + blockIdx.x % (kC - 1);

  if (tid == 0) *cnt = 0;
  for (int w = tid; w < 96; w += 256) sup[w] = 0u;

  // warm GL2 with this batch's boxes (gathered in phase C)  -> global_prefetch_b8
  const float* bbase = boxes + (size_t)b * N * 4;
  for (int n = tid * 8; n < N; n += 256 * 8)
    __builtin_prefetch(bbase + (size_t)n * 4, 0, 1);
  __syncthreads();

  // ---- Phase A: TDM-stage probs chunks into LDS (ping-pong), argmax-gather ----
  // NOTE: scalar (runtime) addrspacecasts only — pointer arrays with shared-
  // derived initializers become unsupported static initializers in ld.lld.
  const float* pb = probs + (size_t)b * N * kC;
  float* stage0 = (float*)(smem + 24064);
  float* stage1 = (float*)(smem + 24064 + kChunk * kC * 4);
  unsigned soff0 = (unsigned)(size_t)stage0;
  unsigned soff1 = (unsigned)(size_t)stage1;
  const int NCH = (N + kChunk - 1) / kChunk;

  if (tid < 32)  // wave 0 owns the TENSORcnt
    tdm_load_probs(soff0, pb, (N < kChunk ? N : kChunk), N);

  for (int c = 0; c < NCH; ++c) {
    const int row0 = c * kChunk;
    const int rows = (N - row0 < kChunk) ? (N - row0) : kChunk;
    if (tid < 32) {
      const int nr0 = row0 + kChunk;
      if (nr0 < N) {  // issue next chunk, then wait for current only (<=1 in flight)
        const int nrows = (N - nr0 < kChunk) ? (N - nr0) : kChunk;
        tdm_load_probs(((c + 1) & 1) ? soff1 : soff0, pb + (size_t)nr0 * kC,
                       nrows, N - nr0);
        __builtin_amdgcn_s_wait_tensorcnt(1);
      } else {
        __builtin_amdgcn_s_wait_tensorcnt(0);
      }
    }
    __syncthreads();
    const float* buf = (c & 1) ? stage1 : stage0;
    for (int r = tid; r < rows; r += 256) {
      const float* row = buf + r * kC;
      float best = row[0];
      int bi = 0;
#pragma unroll
      for (int cc = 1; cc < kC; ++cc) {
        float v = row[cc];
        if (v > best) { best = v; bi = cc; }  // first-max == jnp.argmax
      }
      if (bi == cls) {
        int s = atomicAdd(cnt, 1);
        sc[s] = best;          // score = probs[b][n][cls]
        nidx[s] = row0 + r;
      }
    }
    __syncthreads();
  }
  const int M = *cnt;

  // ---- Phase B: rank by (score desc, index asc) — matches argmax tie-break ----
  for (int i = tid; i < M; i += 256) {
    float si = sc[i];
    int   ni = nidx[i];
    int   r  = 0;
    for (int j = 0; j < M; ++j) {
      float sj = sc[j];
      r += (sj > si) || (sj == si && nidx[j] < ni);
    }
    rk[i] = r;
  }
  __syncthreads();
  for (int i = tid; i < M; i += 256) sidx[rk[i]] = nidx[i];
  __syncthreads();

  // ---- Phase C: async-gather sorted boxes (16B each) straight into LDS ----
  for (int r = tid; r < M; r += 256) {
    int n = sidx[r];
    unsigned lds_off = (unsigned)(size_t)(&bbox[r]);            // LDS byte offset
    unsigned long long ga = (unsigned long long)(size_t)(bbase + (size_t)n * 4);
    asm volatile("global_load_async_to_lds_b128 %0, %1, off"
                 :: "v"(lds_off), "v"(ga) : "memory");
  }
  asm volatile("s_wait_asynccnt 0x0" ::: "memory");
  __syncthreads();

  // ---- Phase D: greedy suppression, 256 lanes vs one leader per step ----
  for (int pos = 0; pos < M; ++pos) {
    bool dead = (sup[pos >> 5] >> (pos & 31)) & 1u;  // uniform across block
    if (!dead) {
      if (tid == 0) keep[(size_t)b * N + sidx[pos]] = 1.0f;
      float4 kb = bbox[pos];
      float ka = fmaxf(kb.z - kb.x, 0.f) * fmaxf(kb.w - kb.y, 0.f);
      for (int j = pos + 1 + tid; j < M; j += 256) {
        float4 cb = bbox[j];
        float ih = fmaxf(fminf(kb.z, cb.z) - fmaxf(kb.x, cb.x), 0.f);
        float iw = fmaxf(fminf(kb.w, cb.w) - fmaxf(kb.y, cb.y), 0.f);
        float inter = ih * iw;
        float ca = fmaxf(cb.z - cb.x, 0.f) * fmaxf(cb.w - cb.y, 0.f);
        float iou = inter / fmaxf(ka + ca - inter, 1e-9f);
        if (iou > 0.5f) atomicOr(&sup[j >> 5], 1u << (j & 31));
      }
    }
    __syncthreads();
  }
}

extern "C" void kernel_launch(void* const* d_in, const int* in_sizes, int n_in,
                              void* d_out, int out_size, void* d_ws, size_t ws_size,
                              hipStream_t stream) {
  const float* nms_reg  = (const float*)d_in[0];
  const float* nms_cls  = (const float*)d_in[1];
  const float* rcnn_reg = (const float*)d_in[2];
  const float* rcnn_cls = (const float*)d_in[3];
  const int*   red      = (const int*)d_in[4];

  const int BN = in_sizes[1];  // B*N = 12000
  const int B  = 4;
  const int N  = BN / B;       // 3000

  float* out      = (float*)d_out;
  float* o_round  = out;
  float* o_sig    = out + (size_t)BN * 4;
  float* o_boxes  = out + (size_t)BN * 5;
  float* o_probs  = out + (size_t)BN * 9;
  float* o_keep   = out + (size_t)BN * 9 + (size_t)BN * kC;

  head_kernel<<<(BN + 255) / 256, 256, 0, stream>>>(
      nms_reg, nms_cls, rcnn_reg, rcnn_cls, red,
      o_round, o_sig, o_boxes, o_probs, o_keep, BN);

  nms_kernel<<<B * (kC - 1), 256, 0, stream>>>(o_boxes, o_probs, o_keep, N);
}